// CenterTOpEXnewMultiC_5617817223886
// MI455X (gfx1250) — compile-verified
//
#include <hip/hip_runtime.h>
#include <math.h>

typedef __attribute__((ext_vector_type(2))) float v2f;
typedef __attribute__((ext_vector_type(8))) float v8f;

#define B_ 4
#define C_ 256
#define N_ 65536
#define K_ 16

// float offsets inside d_out (tuple members, flat, in return order)
#define O_CTR  ((size_t)0)                         // [K,C]
#define O_LAB  ((size_t)K_*C_)                     // [B,N]
#define O_OH   (O_LAB + (size_t)B_*N_)             // [B,N,2]
#define O_W    (O_OH + 2ull*B_*N_)                 // [B,N,2]
#define O_D2C  (O_W + 2ull*B_*N_)                  // [B,N,2]
#define O_LABP (O_D2C + 2ull*B_*N_)                // [B,N]
#define O_CINI (O_LABP + (size_t)B_*N_)            // scalar

// order-preserving float <-> uint keys for atomic min/max
__device__ __forceinline__ unsigned fkey(float f) {
  unsigned u = __float_as_uint(f);
  return (u & 0x80000000u) ? ~u : (u | 0x80000000u);
}
__device__ __forceinline__ float fdec(unsigned k) {
  unsigned u = (k & 0x80000000u) ? (k & 0x7fffffffu) : ~k;
  return __uint_as_float(u);
}

__global__ __launch_bounds__(256) void k_init(float* cacc, int* ccnt, float* gsum,
                                              int* gn, unsigned* gkeys) {
  int i = blockIdx.x * 256 + threadIdx.x;
  if (i < B_ * K_ * C_) cacc[i] = 0.f;
  if (i < B_ * K_) ccnt[i] = 0;
  if (i < B_ * 4) gsum[i] = 0.f;
  if (i < B_) gn[i] = 0;
  if (i < B_ * 4) gkeys[i] = ((i & 1) == 0) ? 0xFFFFFFFFu : 0u;  // even=min, odd=max
}

// normalize centers: cnorm[k][c] = centers[k][c] / max(||centers[k]||, 1e-12)
__global__ __launch_bounds__(256) void k_prep(const float* __restrict__ centers,
                                              float* __restrict__ cnorm) {
  __shared__ float red[256];
  int c = threadIdx.x;
  for (int k = 0; k < K_; ++k) {
    float x = centers[k * C_ + c];
    red[c] = x * x;
    __syncthreads();
    for (int s = 128; s; s >>= 1) { if (c < s) red[c] += red[c + s]; __syncthreads(); }
    float nrm = fmaxf(sqrtf(red[0]), 1e-12f);
    __syncthreads();
    cnorm[k * C_ + c] = x / nrm;
  }
}

// Pass 1: WMMA fp32 distance kernel.  Each wave: 16 pixels x 16 centers,
// C=256 contraction in 64 steps of V_WMMA_F32_16X16X4_F32.
__global__ __launch_bounds__(128) void k_dist(const float* __restrict__ feat,
                                              const float* __restrict__ cnorm,
                                              int* __restrict__ labelsori,
                                              float* __restrict__ out,
                                              float* __restrict__ gsum,
                                              int* __restrict__ gn,
                                              unsigned* __restrict__ gkeys) {
  __shared__ float sC[K_ * (C_ + 1)];  // padded stride: kill bank conflicts
  __shared__ float sSum[4];
  __shared__ int sN;
  __shared__ unsigned sK[4];
  for (int i = threadIdx.x; i < K_ * C_; i += 128)
    sC[(i >> 8) * (C_ + 1) + (i & 255)] = cnorm[i];
  if (threadIdx.x < 4) { sSum[threadIdx.x] = 0.f; sK[threadIdx.x] = (threadIdx.x & 1) ? 0u : 0xFFFFFFFFu; }
  if (threadIdx.x == 0) sN = 0;
  __syncthreads();

  const int b = blockIdx.y;
  const int wave = threadIdx.x >> 5;
  const int lane = threadIdx.x & 31;
  const int half = lane >> 4;          // K-pair selector of the WMMA fragment
  const int m = lane & 15;             // A-row (pixel) and B-col (center)
  const int tile = blockIdx.x * 64 + wave * 16;
  const int px = tile + m;
  const float* fb = feat + (size_t)b * C_ * N_;
  const int crow = m * (C_ + 1);

  v8f acc = {0.f, 0.f, 0.f, 0.f, 0.f, 0.f, 0.f, 0.f};
  float ssq = 0.f;
#pragma unroll 4
  for (int c0 = 0; c0 < C_; c0 += 4) {
    int ca = c0 + 2 * half;
    float a0 = fb[(size_t)ca * N_ + px];        // coalesced: lanes vary over N
    float a1 = fb[(size_t)(ca + 1) * N_ + px];
    ssq = fmaf(a0, a0, ssq);
    ssq = fmaf(a1, a1, ssq);
    v2f A = {a0, a1};                            // A[m, ca..ca+1]
    v2f Bf = {sC[crow + ca], sC[crow + ca + 1]}; // B[ca..ca+1, center m]
    acc = __builtin_amdgcn_wmma_f32_16x16x4_f32(false, A, false, Bf, (short)0,
                                                acc, false, false);
  }
  ssq += __shfl_xor(ssq, 16, 32);                      // combine channel halves
  float rn = 1.0f / fmaxf(sqrtf(ssq), 1e-12f);         // 1/max(||f||,1e-12)

#pragma unroll
  for (int j = 0; j < 8; ++j) {
    int pidx = j + 8 * half;                           // D row = pixel-in-tile
    float r = __shfl(rn, pidx, 32);
    float d = 0.5f * (1.0f - acc[j] * r);              // dist[pixel, center m]
    float su = (m < 8) ? d : 0.f;                      // unchg centers 0..7
    float sc = (m >= 8) ? d : 0.f;                     // chg centers 8..15
    float mv = d; int mi = m;
#pragma unroll
    for (int off = 8; off; off >>= 1) {                // stays within 16-lane half
      su += __shfl_xor(su, off, 32);
      sc += __shfl_xor(sc, off, 32);
      float ov = __shfl_xor(mv, off, 32);
      int oi = __shfl_xor(mi, off, 32);
      if (ov < mv || (ov == mv && oi < mi)) { mv = ov; mi = oi; }  // first-min ties
    }
    if (m == 0) {
      float d0 = su * 0.125f, d1 = sc * 0.125f;
      int lab = mi > (K_ / 2 - 1);
      size_t gi = (size_t)b * N_ + (tile + pidx);
      labelsori[gi] = mi;
      float flab = (float)lab;
      out[O_LAB + gi] = flab;
      out[O_LABP + gi] = flab;
      out[O_OH + 2 * gi] = 1.f - flab;
      out[O_OH + 2 * gi + 1] = flab;
      out[O_D2C + 2 * gi] = d0;
      out[O_D2C + 2 * gi + 1] = d1;
      float th0 = lab ? 0.f : d0;
      float th1 = lab ? d1 : 0.f;
      atomicAdd(&sSum[0], th0); atomicAdd(&sSum[1], th0 * th0);
      atomicAdd(&sSum[2], th1); atomicAdd(&sSum[3], th1 * th1);
      if (lab) atomicAdd(&sN, 1);
      atomicMin(&sK[0], fkey(d0)); atomicMax(&sK[1], fkey(d0));
      atomicMin(&sK[2], fkey(d1)); atomicMax(&sK[3], fkey(d1));
    }
  }
  __syncthreads();
  if (threadIdx.x < 4) atomicAdd(&gsum[b * 4 + threadIdx.x], sSum[threadIdx.x]);
  if (threadIdx.x == 0) atomicAdd(&gn[b], sN);
  if (threadIdx.x == 4) atomicMin(&gkeys[b * 4 + 0], sK[0]);
  if (threadIdx.x == 5) atomicMax(&gkeys[b * 4 + 1], sK[1]);
  if (threadIdx.x == 6) atomicMin(&gkeys[b * 4 + 2], sK[2]);
  if (threadIdx.x == 7) atomicMax(&gkeys[b * 4 + 3], sK[3]);
}

// Pass 2: finalize per-batch statistics -> thresholds + min/max normalizers
__global__ void k_stats(const float* gsum, const int* gn, const unsigned* gkeys,
                        const float* num1, const float* num2, float* stats2) {
  int b = threadIdx.x;
  if (b >= B_) return;
  float s0 = gsum[b * 4 + 0], s0q = gsum[b * 4 + 1];
  float s1 = gsum[b * 4 + 2], s1q = gsum[b * 4 + 3];
  float n1 = (float)gn[b];
  float n0 = (float)(N_ - gn[b]);
  float chg_mean = s1 / (n1 + 1.f);      // reference uses count+1 for means
  float unchg_mean = s0 / (n0 + 1.f);
  float mu1 = s1 / n1, mu0 = s0 / n0;    // variance uses exact count
  float var1 = (s1q - n1 * mu1 * mu1) / (n1 - 1.f);
  float var0 = (s0q - n0 * mu0 * mu0) / (n0 - 1.f);
  stats2[b * 8 + 0] = chg_mean + num1[0] * var1;
  stats2[b * 8 + 1] = unchg_mean + num2[0] * var0;
  float mn0 = fdec(gkeys[b * 4 + 0]), mx0 = fdec(gkeys[b * 4 + 1]);
  float mn1 = fdec(gkeys[b * 4 + 2]), mx1 = fdec(gkeys[b * 4 + 3]);
  stats2[b * 8 + 2] = mn0;
  stats2[b * 8 + 3] = 1.f / (mx0 - mn0 + 1e-7f);
  stats2[b * 8 + 4] = mn1;
  stats2[b * 8 + 5] = 1.f / (mx1 - mn1 + 1e-7f);
}

// Pass 3: weight = 1 - (d2c - mn)/(mx - mn + 1e-7)
__global__ __launch_bounds__(256) void k_weight(const float* __restrict__ stats2,
                                                float* __restrict__ out) {
  size_t i = (size_t)blockIdx.x * 256 + threadIdx.x;  // over B*N
  int b = (int)(i / N_);
  float d0 = out[O_D2C + 2 * i], d1 = out[O_D2C + 2 * i + 1];
  out[O_W + 2 * i] = 1.f - (d0 - stats2[b * 8 + 2]) * stats2[b * 8 + 3];
  out[O_W + 2 * i + 1] = 1.f - (d1 - stats2[b * 8 + 4]) * stats2[b * 8 + 5];
}

// Pass 4: second streaming pass — per-center feature sums via LDS binning
__global__ __launch_bounds__(256) void k_accum(const float* __restrict__ feat,
                                               const int* __restrict__ labelsori,
                                               const float* __restrict__ d2c,
                                               const float* __restrict__ stats2,
                                               float* __restrict__ cacc,
                                               int* __restrict__ ccnt) {
  __shared__ float sAcc[K_ * (C_ + 1)];  // padded stride 257: conflict-free ds_add
  __shared__ int sCnt[K_];
  for (int i = threadIdx.x; i < K_ * (C_ + 1); i += 256) sAcc[i] = 0.f;
  if (threadIdx.x < K_) sCnt[threadIdx.x] = 0;
  __syncthreads();
  const int b = blockIdx.y;
  const int n = blockIdx.x * 256 + threadIdx.x;
  size_t gi = (size_t)b * N_ + n;
  int k = labelsori[gi];
  float d0 = d2c[2 * gi], d1 = d2c[2 * gi + 1];
  int lab = k > (K_ / 2 - 1);
  float th1 = lab ? d1 : 0.f;
  float th0 = lab ? 0.f : d0;
  bool keep = !(th1 > stats2[b * 8 + 0]) && !(th0 > stats2[b * 8 + 1]);
  if (keep) atomicAdd(&sCnt[k], 1);
  const float* fb = feat + (size_t)b * C_ * N_;
  int krow = k * (C_ + 1);
  for (int c = 0; c < C_; ++c) {
    float v = fb[(size_t)c * N_ + n];   // fully coalesced block-wide load
    if (keep) atomicAdd(&sAcc[krow + c], v);
  }
  __syncthreads();
  for (int i = threadIdx.x; i < K_ * C_; i += 256) {
    float v = sAcc[(i >> 8) * (C_ + 1) + (i & 255)];
    if (v != 0.f) atomicAdd(&cacc[(size_t)b * K_ * C_ + i], v);
  }
  if (threadIdx.x < K_) atomicAdd(&ccnt[b * K_ + threadIdx.x], sCnt[threadIdx.x]);
}

// Pass 5: centers_out and cini (cosine similarity of last batch's centers)
__global__ __launch_bounds__(256) void k_final(const float* __restrict__ cacc,
                                               const int* __restrict__ ccnt,
                                               const float* __restrict__ centers,
                                               float* __restrict__ out) {
  __shared__ float r0[256], r1[256], r2[256];
  int c = threadIdx.x;
  for (int k = 0; k < K_; ++k) {
    float s = 0.f;
    for (int b = 0; b < B_; ++b)
      s += cacc[(size_t)b * K_ * C_ + k * C_ + c] / ((float)ccnt[b * K_ + k] + 1.f);
    out[O_CTR + k * C_ + c] = s * (1.f / B_);
  }
  float cini = 0.f;
  for (int k = 0; k < K_; ++k) {
    float ci = cacc[(size_t)(B_ - 1) * K_ * C_ + k * C_ + c] /
               ((float)ccnt[(B_ - 1) * K_ + k] + 1.f);
    float ce = centers[k * C_ + c];
    r0[c] = ci * ce; r1[c] = ci * ci; r2[c] = ce * ce;
    __syncthreads();
    for (int s = 128; s; s >>= 1) {
      if (c < s) { r0[c] += r0[c + s]; r1[c] += r1[c + s]; r2[c] += r2[c + s]; }
      __syncthreads();
    }
    if (c == 0)
      cini += r0[0] / (fmaxf(sqrtf(r1[0]), 1e-8f) * fmaxf(sqrtf(r2[0]), 1e-8f));
    __syncthreads();
  }
  if (c == 0) out[O_CINI] = cini * (1.f / B_);
}

extern "C" void kernel_launch(void* const* d_in, const int* in_sizes, int n_in,
                              void* d_out, int out_size, void* d_ws, size_t ws_size,
                              hipStream_t stream) {
  (void)in_sizes; (void)n_in; (void)out_size; (void)ws_size;
  const float* feat = (const float*)d_in[0];     // [B,C,N]
  const float* centers = (const float*)d_in[1];  // [K,C]
  const float* num1 = (const float*)d_in[2];
  const float* num2 = (const float*)d_in[3];
  float* out = (float*)d_out;

  char* w = (char*)d_ws;
  float* cnorm = (float*)w;        w += (size_t)K_ * C_ * sizeof(float);
  int* labelsori = (int*)w;        w += (size_t)B_ * N_ * sizeof(int);
  float* cacc = (float*)w;         w += (size_t)B_ * K_ * C_ * sizeof(float);
  int* ccnt = (int*)w;             w += (size_t)B_ * K_ * sizeof(int);
  float* gsum = (float*)w;         w += (size_t)B_ * 4 * sizeof(float);
  int* gn = (int*)w;               w += (size_t)B_ * sizeof(int);
  unsigned* gkeys = (unsigned*)w;  w += (size_t)B_ * 4 * sizeof(unsigned);
  float* stats2 = (float*)w;       w += (size_t)B_ * 8 * sizeof(float);

  k_init<<<(B_ * K_ * C_ + 255) / 256, 256, 0, stream>>>(cacc, ccnt, gsum, gn, gkeys);
  k_prep<<<1, 256, 0, stream>>>(centers, cnorm);
  dim3 g1(N_ / 64, B_);
  k_dist<<<g1, 128, 0, stream>>>(feat, cnorm, labelsori, out, gsum, gn, gkeys);
  k_stats<<<1, 32, 0, stream>>>(gsum, gn, gkeys, num1, num2, stats2);
  k_weight<<<(B_ * N_) / 256, 256, 0, stream>>>(stats2, out);
  dim3 g4(N_ / 256, B_);
  k_accum<<<g4, 256, 0, stream>>>(feat, labelsori, out + O_D2C, stats2, cacc, ccnt);
  k_final<<<1, 256, 0, stream>>>(cacc, ccnt, centers, out);
}